// CARROTFeatureAug_64690797412812
// MI455X (gfx1250) — compile-verified
//
#include <hip/hip_runtime.h>
#include <hip/hip_bf16.h>

typedef __attribute__((ext_vector_type(2))) float v2f;
typedef __attribute__((ext_vector_type(4))) float v4f;
typedef __attribute__((ext_vector_type(8))) float v8f;

#define NTHR      256
#define K_TOP     10
#define MROWS     12      // real rows: a_0..a_9, z, noise
#define MROWS_PAD 16      // rows 12..15 zeroed so WMMA A/B reads need no masking
#define MSTRIDE   2052    // 2048 + 4 pad: row stride % 64 banks == 4 -> conflict-free

static constexpr float GAMMA_C  = 0.0f;
static constexpr float EPS_C    = 1e-8f;
static constexpr float QR_EPS_C = 1e-6f;
static constexpr float LOG_TERM = 2.9957322735539909f; // log(1/0.05)

__global__ __launch_bounds__(NTHR) void carrot_aug_kernel(
    const float* __restrict__ zin, const int* __restrict__ yin,
    const float* __restrict__ logits, const float* __restrict__ Wm,
    const float* __restrict__ noise, float* __restrict__ out,
    int B, int D, int C)
{
  extern __shared__ float smem[];
  float* Mlds  = smem;                               // MROWS_PAD * MSTRIDE
  float* S     = Mlds + MROWS_PAD * MSTRIDE;         // 256 floats (16x16 Gram)
  float* cval  = S + 256;                            // NTHR*K_TOP candidate values
  int*   cidx  = (int*)(cval + NTHR * K_TOP);        // NTHR*K_TOP candidate indices
  float* rval  = (float*)(cidx + NTHR * K_TOP);      // 8 per-wave reduce values
  int*   rslot = (int*)(rval + NTHR);                // 8 per-wave reduce slots
  int*   topk  = rslot + NTHR;                       // K_TOP winning class ids
  float* bc    = (float*)(topk + K_TOP);             // 16 floats: beta[10], scale, valid

  const int b    = blockIdx.x;
  const int tid  = threadIdx.x;
  const int lane = tid & 31;
  const int ycls = yin[b];
  const float* lrow = logits + (size_t)b * C;
  const float NEG_INF = -__builtin_inff();
  const float POS_INF =  __builtin_inff();

  // ---------- Phase 0: per-thread local top-K over strided logits ----------
  float tv[K_TOP]; int ti[K_TOP];
#pragma unroll
  for (int j = 0; j < K_TOP; ++j) { tv[j] = NEG_INF; ti[j] = 0; }
  for (int c = tid; c < C; c += NTHR) {
    if (c == ycls) continue;                 // exclude true class
    float v = lrow[c];
    if (v > tv[K_TOP - 1]) {
      tv[K_TOP - 1] = v; ti[K_TOP - 1] = c;  // insert + bubble up (array sorted desc)
#pragma unroll
      for (int j = K_TOP - 1; j > 0; --j) {
        if (tv[j] > tv[j - 1]) {
          float t = tv[j]; tv[j] = tv[j - 1]; tv[j - 1] = t;
          int   q = ti[j]; ti[j] = ti[j - 1]; ti[j - 1] = q;
        }
      }
    }
  }
#pragma unroll
  for (int j = 0; j < K_TOP; ++j) {
    cval[tid * K_TOP + j] = tv[j];
    cidx[tid * K_TOP + j] = ti[j];
  }
  __syncthreads();

  // ---------- Merge: K rounds, wave-level shuffle reduce + 8-way cross-wave ----------
  for (int r = 0; r < K_TOP; ++r) {
    float best = NEG_INF; int bslot = -1;
    for (int s = tid; s < NTHR * K_TOP; s += NTHR) {
      float v = cval[s];
      if (v > best) { best = v; bslot = s; }
    }
#pragma unroll
    for (int off = 16; off > 0; off >>= 1) {         // intra-wave argmax, no barriers
      float ov = __shfl_down(best, off, 32);
      int   os = __shfl_down(bslot, off, 32);
      if (ov > best) { best = ov; bslot = os; }
    }
    if (lane == 0) { rval[tid >> 5] = best; rslot[tid >> 5] = bslot; }
    __syncthreads();
    if (tid == 0) {
      float wb = rval[0]; int ws = rslot[0];
#pragma unroll
      for (int w = 1; w < NTHR / 32; ++w) {
        if (rval[w] > wb) { wb = rval[w]; ws = rslot[w]; }
      }
      topk[r] = (ws >= 0) ? cidx[ws] : 0;
      if (ws >= 0) cval[ws] = NEG_INF;
    }
    __syncthreads();
  }

  // ---------- Phase 1: build M rows in LDS: a_0..a_9, z, noise, zeros ----------
  const float* wy = Wm + (size_t)ycls * D;
  for (int k = 0; k < K_TOP; ++k) {
    const float* wk = Wm + (size_t)topk[k] * D;
    for (int d = tid * 4; d < D; d += NTHR * 4) {
      __builtin_prefetch(wk + d + NTHR * 4, 0, 0);   // global_prefetch_b8
      v4f a  = *(const v4f*)(wy + d);
      v4f bb = *(const v4f*)(wk + d);
      *(v4f*)(Mlds + k * MSTRIDE + d) = a - bb;
    }
  }
  {
    const v4f zero4 = {0.0f, 0.0f, 0.0f, 0.0f};
    for (int d = tid * 4; d < D; d += NTHR * 4) {
      *(v4f*)(Mlds + 10 * MSTRIDE + d) = *(const v4f*)(zin   + (size_t)b * D + d);
      *(v4f*)(Mlds + 11 * MSTRIDE + d) = *(const v4f*)(noise + (size_t)b * D + d);
      *(v4f*)(Mlds + 12 * MSTRIDE + d) = zero4;
      *(v4f*)(Mlds + 13 * MSTRIDE + d) = zero4;
      *(v4f*)(Mlds + 14 * MSTRIDE + d) = zero4;
      *(v4f*)(Mlds + 15 * MSTRIDE + d) = zero4;
    }
  }
  S[tid & 255] = 0.0f;   // zero the 16x16 Gram accumulator
  __syncthreads();

  // ---------- Phase 2: Gram S = Mext * Mext^T via V_WMMA_F32_16X16X4_F32 ----------
  // A operand 16x4 f32: lane L holds row (L&15), k-pair selected by (L>>4).
  // B operand (4x16 slice of M^T) mirrors the A layout per-lane, so B == A.
  {
    // readfirstlane -> scalar wave id -> uniform sgpr loop (EXEC stays all-1s)
    const int wave = __builtin_amdgcn_readfirstlane(tid >> 5);
    const int row  = lane & 15;
    const int half = lane >> 4;
    const float* rp = Mlds + row * MSTRIDE + 2 * half;
    const int dchunk = D >> 3;             // 256 columns per wave
    const int d_beg = wave * dchunk;
    const int d_end = d_beg + dchunk;
    v8f acc = {};
    for (int d0 = d_beg; d0 < d_end; d0 += 16) {     // 4 WMMAs per iteration
      v2f a0 = *(const v2f*)(rp + d0);
      v2f a1 = *(const v2f*)(rp + d0 + 4);
      v2f a2 = *(const v2f*)(rp + d0 + 8);
      v2f a3 = *(const v2f*)(rp + d0 + 12);
      acc = __builtin_amdgcn_wmma_f32_16x16x4_f32(false, a0, false, a0, (short)0, acc, false, false);
      acc = __builtin_amdgcn_wmma_f32_16x16x4_f32(false, a1, false, a1, (short)0, acc, false, false);
      acc = __builtin_amdgcn_wmma_f32_16x16x4_f32(false, a2, false, a2, (short)0, acc, false, false);
      acc = __builtin_amdgcn_wmma_f32_16x16x4_f32(false, a3, false, a3, (short)0, acc, false, false);
    }
    // C/D layout: lane 0-15 -> N=lane, M=i ; lane 16-31 -> N=lane-16, M=i+8
#pragma unroll
    for (int i = 0; i < 8; ++i) {
      atomicAdd(&S[(i + 8 * half) * 16 + row], acc[i]);   // ds_add_f32 merge across waves
    }
  }
  __syncthreads();

  // ---------- Phase 3: coefficient-space rank-filtered Gram-Schmidt (one lane) ----------
  if (tid == 0) {
    float G[K_TOP][K_TOP], gz[K_TOP], gn[K_TOP];
    for (int i = 0; i < K_TOP; ++i) {
      for (int j = 0; j < K_TOP; ++j) G[i][j] = S[i * 16 + j];
      gz[i] = S[i * 16 + 10];     // a_i . z  (margins)
      gn[i] = S[i * 16 + 11];     // a_i . noise
    }
    // q_r = sum_m cq[r][m] * a_m ; dropped directions get cq[r][*] = 0 (matches Qm)
    float cq[K_TOP][K_TOP];
    bool anykeep = false;
    for (int r = 0; r < K_TOP; ++r) {
      float w[K_TOP];
      for (int m = 0; m < K_TOP; ++m) w[m] = 0.0f;
      w[r] = 1.0f;
      for (int j = 0; j < r; ++j) {
        float t = 0.0f;
        for (int m = 0; m < K_TOP; ++m) t += cq[j][m] * G[m][r];   // q_j . a_r
        for (int m = 0; m < K_TOP; ++m) w[m] -= t * cq[j][m];
      }
      float n2 = 0.0f;
      for (int p = 0; p < K_TOP; ++p) {
        float s2 = 0.0f;
        for (int q = 0; q < K_TOP; ++q) s2 += G[p][q] * w[q];
        n2 += w[p] * s2;
      }
      float rd = __builtin_sqrtf(fmaxf(n2, 0.0f));
      bool keep = rd > QR_EPS_C;
      if (keep) {
        anykeep = true;
        float inv = 1.0f / rd;
        for (int m = 0; m < K_TOP; ++m) cq[r][m] = w[m] * inv;
      } else {
        for (int m = 0; m < K_TOP; ++m) cq[r][m] = 0.0f;
      }
    }
    // alpha = min_k slack^2 / (2 * pn2 * log(1/delta) + eps)
    float alpha = POS_INF;
    for (int k = 0; k < K_TOP; ++k) {
      float proj = 0.0f;
      for (int r = 0; r < K_TOP; ++r) {
        float qa = 0.0f;
        for (int m = 0; m < K_TOP; ++m) qa += cq[r][m] * G[m][k];  // q_r . a_k
        proj += qa * qa;
      }
      float pn2   = fmaxf(G[k][k] - proj, EPS_C);
      float slack = fmaxf(gz[k] - GAMMA_C, 0.0f);
      float ak    = slack * slack / (2.0f * pn2 * LOG_TERM + EPS_C);
      alpha = fminf(alpha, ak);
    }
    bool validb = anykeep && (alpha > 0.0f) && (alpha < POS_INF) && (alpha == alpha);
    float scale = validb ? __builtin_sqrtf(fmaxf(alpha, 0.0f)) : 0.0f;
    // epsN = noise - sum_j beta_j a_j ; beta_j = sum_r (q_r . noise) cq[r][j]
    float beta[K_TOP];
    for (int j = 0; j < K_TOP; ++j) beta[j] = 0.0f;
    for (int r = 0; r < K_TOP; ++r) {
      float nc = 0.0f;
      for (int m = 0; m < K_TOP; ++m) nc += cq[r][m] * gn[m];
      for (int j = 0; j < K_TOP; ++j) beta[j] += nc * cq[r][j];
    }
    for (int j = 0; j < K_TOP; ++j) bc[j] = beta[j];
    bc[10] = scale;
    bc[11] = validb ? 1.0f : 0.0f;
    // tail outputs: y then valid, appended after the (B,D) block
    out[(size_t)B * D + b]     = (float)ycls;
    out[(size_t)B * D + B + b] = validb ? 1.0f : 0.0f;
  }
  __syncthreads();

  // ---------- Phase 4: z_aug = z + scale * (noise - sum_j beta_j a_j) ----------
  float sb[K_TOP];
#pragma unroll
  for (int j = 0; j < K_TOP; ++j) sb[j] = bc[j];
  const float scale = bc[10];
  float* orow = out + (size_t)b * D;
  for (int d = tid * 4; d < D; d += NTHR * 4) {
    v4f zv = *(const v4f*)(Mlds + 10 * MSTRIDE + d);
    v4f nv = *(const v4f*)(Mlds + 11 * MSTRIDE + d);
    v4f acc4 = nv;
#pragma unroll
    for (int j = 0; j < K_TOP; ++j) {
      v4f av = *(const v4f*)(Mlds + j * MSTRIDE + d);
      acc4 -= sb[j] * av;
    }
    *(v4f*)(orow + d) = zv + scale * acc4;
  }
}

extern "C" void kernel_launch(void* const* d_in, const int* in_sizes, int n_in,
                              void* d_out, int out_size, void* d_ws, size_t ws_size,
                              hipStream_t stream) {
  const float* z      = (const float*)d_in[0];
  const int*   y      = (const int*)d_in[1];
  const float* logits = (const float*)d_in[2];
  const float* Wm     = (const float*)d_in[3];
  const float* noise  = (const float*)d_in[4];
  float* out = (float*)d_out;

  const int B = in_sizes[1];
  const int D = in_sizes[0] / B;
  const int C = in_sizes[2] / B;

  const size_t smem_elems = (size_t)MROWS_PAD * MSTRIDE // M (16 rows, 12 live)
                          + 256                         // Gram 16x16
                          + (size_t)NTHR * K_TOP * 2    // cval + cidx
                          + (size_t)NTHR * 2            // rval + rslot
                          + K_TOP + 16;                 // topk + broadcast
  const size_t smem_bytes = smem_elems * sizeof(float);

  hipFuncSetAttribute((const void*)carrot_aug_kernel,
                      hipFuncAttributeMaxDynamicSharedMemorySize,
                      (int)smem_bytes);

  carrot_aug_kernel<<<B, NTHR, smem_bytes, stream>>>(z, y, logits, Wm, noise, out, B, D, C);
}